// TPWNG_66846870995235
// MI455X (gfx1250) — compile-verified
//
#include <hip/hip_runtime.h>
#include <math.h>

typedef __attribute__((ext_vector_type(2))) float v2f;
typedef __attribute__((ext_vector_type(4))) float v4f;
typedef __attribute__((ext_vector_type(8))) float v8f;

// ---- problem constants ----
#define DDIM   512
#define KCLS   14
#define NROWS  65536L          // 16 batches * 4096 features
#define FDIM   4096
#define NDIR   16              // 13 anomaly + T_n + w_score + pad

// output offsets (flat concatenation, reference return order)
#define OFF_SNN      0L
#define OFF_PHINA    (OFF_SNN + NROWS)
#define OFF_SAN      (OFF_PHINA + NROWS * 13)
#define OFF_SAATRUE  (OFF_SAN + NROWS)
#define OFF_PHIOTHER (OFF_SAATRUE + NROWS)
#define OFF_SAANORM  (OFF_PHIOTHER + NROWS * 12)
#define OFF_PSEUDO   (OFF_SAANORM + NROWS)
#define OFF_SCORES   (OFF_PSEUDO + NROWS)
#define OFF_TN       (OFF_SCORES + NROWS)

#define KCHUNK     32
#define NCHUNK     (DDIM / KCHUNK)     // 16
#define LDS_STRIDE 36                  // 32 + 4 pad floats (conflict-free A reads)
#define WAVES      8

// ---------------------------------------------------------------------------
// Kernel 1: normalize T rows, build dirs[512][16], write T_n_enhanced
// ---------------------------------------------------------------------------
__global__ void __launch_bounds__(512)
prep_dirs(const float* __restrict__ T_emb, const float* __restrict__ w_score,
          float* __restrict__ dirs, float* __restrict__ out_Tn) {
  __shared__ float snorm[KCLS];
  int tid  = threadIdx.x;
  int wave = tid >> 5, lane = tid & 31;
  if (wave < KCLS) {
    float s = 0.f;
    #pragma unroll
    for (int i = 0; i < DDIM / 32; ++i) {
      float v = T_emb[wave * DDIM + i * 32 + lane];
      s += v * v;
    }
    #pragma unroll
    for (int off = 16; off; off >>= 1) s += __shfl_xor(s, off, 32);
    if (lane == 0) snorm[wave] = sqrtf(s) + 1e-8f;
  }
  __syncthreads();
  int k = tid;  // 0..511
  #pragma unroll
  for (int n = 0; n < NDIR; ++n) {
    float v;
    if (n < KCLS)       v = T_emb[n * DDIM + k] / snorm[n];
    else if (n == KCLS) v = w_score[k];
    else                v = 0.f;
    dirs[k * NDIR + n] = v;
  }
  out_Tn[k] = T_emb[(KCLS - 1) * DDIM + k] / snorm[KCLS - 1];
}

// ---------------------------------------------------------------------------
// Kernel 2: streaming WMMA pass, async global->LDS double-buffered staging.
//   One wave per 16-row tile: P[16][16] = feats_tile(16x512) @ dirs(512x16).
// ---------------------------------------------------------------------------
__device__ __forceinline__ void stage_async(float* buf, const float* A,
                                            int c, int lane) {
  // 16 rows x KCHUNK floats = 128 float4 -> 4 async b128 copies per lane
  #pragma unroll
  for (int i = 0; i < 4; ++i) {
    int flat = i * 32 + lane;            // 0..127 float4 index
    int r = flat >> 3;                   // KCHUNK/4 == 8 float4 per row
    int q = flat & 7;
    unsigned     ldsa = (unsigned)(uintptr_t)(buf + r * LDS_STRIDE + q * 4);
    const float* ga   = A + (long)r * DDIM + c + q * 4;
    asm volatile("global_load_async_to_lds_b128 %0, %1, off"
                 :: "v"(ldsa), "v"(ga) : "memory");
  }
}

__global__ void __launch_bounds__(256)
fused_gemm(const float* __restrict__ feats, const float* __restrict__ dirs,
           const int* __restrict__ cls_idx, float* __restrict__ dout,
           const float* __restrict__ b_score_ptr, int is_anom) {
  __shared__ __align__(16) float dirs_lds[DDIM * NDIR];                 // 32 KB
  __shared__ __align__(16) float stage[WAVES * 2 * 16 * LDS_STRIDE];    // ~36 KB
  const int wave = threadIdx.x >> 5;
  const int lane = threadIdx.x & 31;
  const int half = lane >> 4;
  const int n16  = lane & 15;

  float* buf0 = stage + wave * 2 * 16 * LDS_STRIDE;
  float* buf1 = buf0 + 16 * LDS_STRIDE;

  const long tileIdx = (long)blockIdx.x * WAVES + wave;  // 0..4095
  const long row0    = tileIdx * 16;
  const float* A     = feats + row0 * DDIM;

  // cooperative load of dirs (512x16 f32) into LDS, b128 coalesced
  for (int i = threadIdx.x; i < DDIM * NDIR / 4; i += 256) {
    *(v4f*)(dirs_lds + i * 4) = *(const v4f*)(dirs + i * 4);
  }

  // prime the async pipeline with chunk 0
  stage_async(buf0, A, 0, lane);
  __syncthreads();   // dirs_lds visible to all waves

  v8f acc = {};

  for (int cidx = 0; cidx < NCHUNK; ++cidx) {
    const int c = cidx * KCHUNK;
    float* cur = (cidx & 1) ? buf1 : buf0;
    float* nxt = (cidx & 1) ? buf0 : buf1;

    if (cidx + 1 < NCHUNK) {
      // prior DS reads of 'nxt' must drain before async engine overwrites it
      asm volatile("s_wait_dscnt 0x0" ::: "memory");
      stage_async(nxt, A, c + KCHUNK, lane);
      // async loads complete in order: <=4 outstanding => chunk cidx arrived
      asm volatile("s_wait_asynccnt 0x4" ::: "memory");
    } else {
      asm volatile("s_wait_asynccnt 0x0" ::: "memory");
    }

    #pragma unroll
    for (int kk = 0; kk < KCHUNK; kk += 4) {
      // A fragment (16x4 f32): lane<16 -> K=kk,kk+1 ; lane>=16 -> K=kk+2,kk+3
      v2f a = *(const v2f*)(cur + n16 * LDS_STRIDE + kk + 2 * half);
      // B fragment (4x16 f32) from LDS-resident dirs
      int kg = c + kk + 2 * half;
      v2f b;
      b.x = dirs_lds[kg * NDIR + n16];
      b.y = dirs_lds[(kg + 1) * NDIR + n16];
      acc = __builtin_amdgcn_wmma_f32_16x16x4_f32(
          /*neg_a=*/false, a, /*neg_b=*/false, b,
          /*c_mod=*/(short)0, acc, /*reuse_a=*/false, /*reuse_b=*/false);
    }
  }

  // dump 16x16 tile to LDS (stride 17) so global writes can be coalesced
  float* pt = buf0;
  #pragma unroll
  for (int j = 0; j < 8; ++j) {
    int m = j + half * 8;                 // C/D layout: vgpr j, lane half
    pt[m * 17 + n16] = acc[j];
  }

  const long g = row0;
  if (!is_anom) {
    if (half == 0) dout[OFF_SNN + g + n16] = pt[n16 * 17 + 13];
    for (int t = lane; t < 16 * 13; t += 32) {
      int r = t / 13, cd = t % 13;
      dout[OFF_PHINA + g * 13 + t] = pt[r * 17 + cd];
    }
  } else {
    const int b   = (int)(g >> 12);       // batch (tiles never cross batches)
    const int cls = cls_idx[b];
    const float bsc = *b_score_ptr;
    if (half == 0) {
      dout[OFF_SAN + g + n16]     = pt[n16 * 17 + 13];
      dout[OFF_SAATRUE + g + n16] = pt[n16 * 17 + cls];
    } else {
      float logit = pt[n16 * 17 + 14] + bsc;
      dout[OFF_SCORES + g + n16] = 1.f / (1.f + __expf(-logit));
    }
    for (int t = lane; t < 16 * 12; t += 32) {
      int r = t / 12, j = t % 12;
      int cd = (j < cls) ? j : j + 1;
      dout[OFF_PHIOTHER + g * 12 + t] = pt[r * 17 + cd];
    }
  }
}

// ---------------------------------------------------------------------------
// Kernel 3: per-batch min/max of S_aa_true and S_an
// ---------------------------------------------------------------------------
__global__ void __launch_bounds__(256)
batch_minmax(const float* __restrict__ dout, float* __restrict__ mm) {
  __shared__ float sA0[256], sA1[256], sN0[256], sN1[256];
  const int b = blockIdx.x, tid = threadIdx.x;
  float mnA = 3.4e38f, mxA = -3.4e38f, mnN = 3.4e38f, mxN = -3.4e38f;
  for (int f = tid; f < FDIM; f += 256) {
    float a = dout[OFF_SAATRUE + (long)b * FDIM + f];
    float n = dout[OFF_SAN + (long)b * FDIM + f];
    mnA = fminf(mnA, a); mxA = fmaxf(mxA, a);
    mnN = fminf(mnN, n); mxN = fmaxf(mxN, n);
  }
  sA0[tid] = mnA; sA1[tid] = mxA; sN0[tid] = mnN; sN1[tid] = mxN;
  __syncthreads();
  for (int s = 128; s > 0; s >>= 1) {
    if (tid < s) {
      sA0[tid] = fminf(sA0[tid], sA0[tid + s]);
      sA1[tid] = fmaxf(sA1[tid], sA1[tid + s]);
      sN0[tid] = fminf(sN0[tid], sN0[tid + s]);
      sN1[tid] = fmaxf(sN1[tid], sN1[tid + s]);
    }
    __syncthreads();
  }
  if (tid == 0) {
    mm[b * 4 + 0] = sA0[0]; mm[b * 4 + 1] = sA1[0];
    mm[b * 4 + 2] = sN0[0]; mm[b * 4 + 3] = sN1[0];
  }
}

// ---------------------------------------------------------------------------
// Kernel 4: minmax-normalize, fuse, threshold
// ---------------------------------------------------------------------------
__global__ void __launch_bounds__(256)
finalize(float* __restrict__ dout, const float* __restrict__ mm) {
  long g = (long)blockIdx.x * 256 + threadIdx.x;
  if (g >= NROWS) return;
  int b = (int)(g >> 12);
  float mnA = mm[b * 4 + 0], mxA = mm[b * 4 + 1];
  float mnN = mm[b * 4 + 2], mxN = mm[b * 4 + 3];
  float saa = (dout[OFF_SAATRUE + g] - mnA) / (mxA - mnA + 1e-8f);
  float san = (dout[OFF_SAN + g] - mnN) / (mxN - mnN + 1e-8f);
  float fused = 0.8f * saa + 0.2f * (1.f - san);
  dout[OFF_SAANORM + g] = saa;
  dout[OFF_PSEUDO + g]  = (fused > 0.55f) ? 1.f : 0.f;
}

// ---------------------------------------------------------------------------
extern "C" void kernel_launch(void* const* d_in, const int* in_sizes, int n_in,
                              void* d_out, int out_size, void* d_ws, size_t ws_size,
                              hipStream_t stream) {
  const float* normal_feats  = (const float*)d_in[0];
  const float* anomaly_feats = (const float*)d_in[1];
  const int*   cls_idx       = (const int*)d_in[2];
  const float* T_emb         = (const float*)d_in[3];
  const float* w_score       = (const float*)d_in[4];
  const float* b_score       = (const float*)d_in[5];
  float* dout = (float*)d_out;

  float* dirs = (float*)d_ws;                          // 512*16 f32 = 32KB
  float* mm   = (float*)((char*)d_ws + 512 * 16 * 4);  // 16*4 f32

  prep_dirs<<<1, 512, 0, stream>>>(T_emb, w_score, dirs, dout + OFF_TN);

  const int blocks = (int)(NROWS / 16 / WAVES);        // 512
  fused_gemm<<<blocks, 256, 0, stream>>>(normal_feats,  dirs, cls_idx, dout, b_score, 0);
  fused_gemm<<<blocks, 256, 0, stream>>>(anomaly_feats, dirs, cls_idx, dout, b_score, 1);

  batch_minmax<<<16, 256, 0, stream>>>(dout, mm);
  finalize<<<(int)(NROWS / 256), 256, 0, stream>>>(dout, mm);
}